// Probe_12790412607931
// MI455X (gfx1250) — compile-verified
//
#include <hip/hip_runtime.h>
#include <math.h>

// Problem constants (match reference: B=2048, T=8192, C=8)
constexpr int Bn = 2048;
constexpr int Tn = 8192;
constexpr int Cn = 8;
constexpr int NCH = 8;                       // T-chunks per batch row
constexpr int Tc = Tn / NCH;                 // 1024 rows per chunk
constexpr int THREADS = 256;                 // 8 waves (wave32)
constexpr int F4_PER_CHUNK = (Tc * Cn) / 4;  // 2048 float4s per chunk
constexpr int ITERS = F4_PER_CHUNK / THREADS;// 8 iterations per thread

typedef float v2f __attribute__((ext_vector_type(2)));
typedef float v8f __attribute__((ext_vector_type(8)));

// ---------------------------------------------------------------------------
// Phase 1: stream x, reduce each (batch, T-chunk) to 8 channels x 3 raw stats
//   stat k=0: count of (x>0)   (a_sum; a_max = count>0)
//   stat k=1: sum  of log1p(x) (l_mean = sum/T)
//   stat k=2: max  of log1p(x) (l_max)
// ws layout: [B][NCH][C][3] floats
// ---------------------------------------------------------------------------
__global__ __launch_bounds__(THREADS) void stats_phase1(
    const float* __restrict__ x, float* __restrict__ ws) {
  const int b   = blockIdx.x / NCH;
  const int ch  = blockIdx.x % NCH;
  const size_t base = ((size_t)b * Tn + (size_t)ch * Tc) * Cn;
  const float4* __restrict__ p = (const float4*)(x + base);
  const int tid = threadIdx.x;

  // tid&1 fixes this thread's channel group: 0 -> ch 0..3, 1 -> ch 4..7
  float cnt[4]  = {0.f, 0.f, 0.f, 0.f};
  float lsum[4] = {0.f, 0.f, 0.f, 0.f};
  float lmax[4] = {-INFINITY, -INFINITY, -INFINITY, -INFINITY};

#pragma unroll
  for (int it = 0; it < ITERS; ++it) {
    if (it + 2 < ITERS)  // gfx1250 global_prefetch_b8
      __builtin_prefetch((const void*)(p + tid + (it + 2) * THREADS), 0, 0);
    float4 v = p[tid + it * THREADS];  // global_load_b128
    float e0[4] = {v.x, v.y, v.z, v.w};
#pragma unroll
    for (int j = 0; j < 4; ++j) {
      cnt[j] += (e0[j] > 0.0f) ? 1.0f : 0.0f;
      float lg = __logf(1.0f + e0[j]);  // v_log_f32; 1+x in [1,2): no cancellation
      lsum[j] += lg;
      lmax[j]  = fmaxf(lmax[j], lg);
    }
  }

  // wave32 xor-reduce with even masks: preserves lane parity (channel group)
#pragma unroll
  for (int m = 16; m >= 2; m >>= 1) {
#pragma unroll
    for (int j = 0; j < 4; ++j) {
      cnt[j]  += __shfl_xor(cnt[j],  m, 32);
      lsum[j] += __shfl_xor(lsum[j], m, 32);
      lmax[j]  = fmaxf(lmax[j], __shfl_xor(lmax[j], m, 32));
    }
  }

  __shared__ float part[8][2][12];  // [wave][group][j*3+k]
  const int wave = tid >> 5;
  const int lane = tid & 31;
  if (lane < 2) {
#pragma unroll
    for (int j = 0; j < 4; ++j) {
      part[wave][lane][j * 3 + 0] = cnt[j];
      part[wave][lane][j * 3 + 1] = lsum[j];
      part[wave][lane][j * 3 + 2] = lmax[j];
    }
  }
  __syncthreads();

  if (tid < 24) {  // 2 groups x 4 ch x 3 stats
    const int g = tid / 12, idx = tid % 12;
    const int j = idx / 3,  k   = idx % 3;
    float v = part[0][g][idx];
#pragma unroll
    for (int w = 1; w < 8; ++w) {
      float o = part[w][g][idx];
      v = (k == 2) ? fmaxf(v, o) : (v + o);
    }
    const int c = g * 4 + j;
    ws[(((size_t)b * NCH + ch) * Cn + c) * 3 + k] = v;
  }
}

// ---------------------------------------------------------------------------
// Phase 2: one wave per 16 batch rows. Fold chunk partials -> 16x32 feature
// tile in LDS, then compute out[b] = feats . W + bias with
// V_WMMA_F32_16X16X4_F32 (B = W broadcast across columns; every D column
// equals the dot).
// ---------------------------------------------------------------------------
__global__ __launch_bounds__(32) void dot_phase2(
    const float* __restrict__ ws, const float* __restrict__ W,
    const float* __restrict__ bias, float* __restrict__ out) {
  __shared__ float raw[16][24];   // [row][c*3+k]
  __shared__ float feat[16][32];  // [row][4*c+s]
  const int lane = threadIdx.x;
  const int b0 = blockIdx.x * 16;

  // Fold NCH chunk-partials: 16 rows * 24 stats = 384 tasks = 32 lanes * 12
#pragma unroll
  for (int i = 0; i < 12; ++i) {
    const int id = lane + 32 * i;
    const int r = id / 24, rem = id % 24;
    const int k = rem % 3;
    const size_t base = ((size_t)(b0 + r) * NCH) * (Cn * 3) + rem % 3 + (rem / 3) * 3;
    float v = ws[base];
#pragma unroll
    for (int ch = 1; ch < NCH; ++ch) {
      float o = ws[base + (size_t)ch * (Cn * 3)];
      v = (k == 2) ? fmaxf(v, o) : (v + o);
    }
    raw[r][rem] = v;
  }
  __syncthreads();

  // Build features: lane owns column = lane -> (c = lane/4, s = lane%4)
  const int c = lane >> 2, s = lane & 3;
#pragma unroll
  for (int r = 0; r < 16; ++r) {
    const float cntv  = raw[r][c * 3 + 0];
    const float lsumv = raw[r][c * 3 + 1];
    const float lmaxv = raw[r][c * 3 + 2];
    float f;
    if      (s == 0) f = (cntv > 0.0f) ? 1.0f : 0.0f;  // a_max
    else if (s == 1) f = cntv;                          // a_sum
    else if (s == 2) f = lmaxv;                         // l_max
    else             f = lsumv * (1.0f / (float)Tn);    // l_mean
    feat[r][lane] = f;
  }
  __syncthreads();

  // WMMA f32 16x16x4, accumulate over 8 K-chunks of 4.
  // A layout: lanes 0-15 -> M=lane, VGPR{0,1}=K{0,1}; lanes 16-31 -> K{2,3}.
  // B gets W[k] replicated to all 16 columns, so D[m,n] = dot(feat[m,:], W).
  const int row   = lane & 15;
  const int khalf = lane >> 4;
  v8f acc = {};
#pragma unroll
  for (int kc = 0; kc < 8; ++kc) {
    const int kbase = kc * 4 + 2 * khalf;
    v2f a, bm;
    a.x  = feat[row][kbase];
    a.y  = feat[row][kbase + 1];
    bm.x = W[kbase];
    bm.y = W[kbase + 1];
    acc = __builtin_amdgcn_wmma_f32_16x16x4_f32(
        /*neg_a=*/false, a, /*neg_b=*/false, bm,
        /*c_mod=*/(short)0, acc, /*reuse_a=*/false, /*reuse_b=*/false);
  }

  // D layout: VGPR j, lane 0 -> (M=j, N=0); lane 16 -> (M=8+j, N=0)
  const float bv = bias[0];
  if (lane == 0) {
#pragma unroll
    for (int j = 0; j < 8; ++j) out[b0 + j] = acc[j] + bv;
  } else if (lane == 16) {
#pragma unroll
    for (int j = 0; j < 8; ++j) out[b0 + 8 + j] = acc[j] + bv;
  }
}

extern "C" void kernel_launch(void* const* d_in, const int* in_sizes, int n_in,
                              void* d_out, int out_size, void* d_ws, size_t ws_size,
                              hipStream_t stream) {
  (void)in_sizes; (void)n_in; (void)out_size; (void)ws_size;
  const float* x    = (const float*)d_in[0];  // [B, T, C]
  const float* W    = (const float*)d_in[1];  // [1, 4C] = 32 floats
  const float* bias = (const float*)d_in[2];  // [1]
  float* out = (float*)d_out;                 // [B] floats
  float* ws  = (float*)d_ws;                  // B*NCH*C*3 floats = 1.5 MiB

  stats_phase1<<<Bn * NCH, THREADS, 0, stream>>>(x, ws);
  dot_phase2<<<Bn / 16, 32, 0, stream>>>(ws, W, bias, out);
}